// GNNActorCritic_40845138985372
// MI455X (gfx1250) — compile-verified
//
#include <hip/hip_runtime.h>
#include <hip/hip_bf16.h>
#include <math.h>

#define N_NODES 16384
#define N_EDGES 524288
#define HID     16
#define KB      64            // K-block for the big GEMM

typedef __attribute__((ext_vector_type(2))) float v2f;
typedef __attribute__((ext_vector_type(8))) float v8f;

// ---------------------------------------------------------------------------
// Async copy of one 4 KB B-block (64 x 16 f32) into LDS: 8 x 512B coalesced
// global_load_async_to_lds_b128 ops, tracked by ASYNCcnt.
// ---------------------------------------------------------------------------
__device__ __forceinline__ void async_copy_bblk(unsigned lds_base,
                                                const float* __restrict__ gsrc,
                                                int lane) {
  const float* g = gsrc + lane * 4;          // 16B per lane per op
  unsigned     l = lds_base + lane * 16;
  #pragma unroll
  for (int t = 0; t < 8; ++t) {
    asm volatile("global_load_async_to_lds_b128 %0, %1, off"
                 :: "v"(l + t * 512), "v"(g + t * 128)
                 : "memory");
  }
}

// ---------------------------------------------------------------------------
// Big WMMA GEMM: C[N x 16] = x[N x N] @ W1[N x 16].
// One wave per 16-row tile. A streams from HBM (16 b64 loads issued per
// 64-K block before consumption), B double-buffered in LDS via async copies.
// HBM-bound: 1.07 GB / 23.3 TB/s ~= 46 us floor; f32 WMMA is exact and free.
// ---------------------------------------------------------------------------
__global__ __launch_bounds__(32) void k_gemm_x(const float* __restrict__ A,
                                               const float* __restrict__ B,
                                               float* __restrict__ C) {
  __shared__ float bs[2][KB * HID];          // 2 x 4 KB double buffer
  const int lane = threadIdx.x;              // 0..31
  const int m    = lane & 15;                // A row in tile / B,D column
  const int h    = lane >> 4;                // K-half selector
  const int rb   = blockIdx.x * 16;
  const float* __restrict__ arow = A + (long)(rb + m) * N_NODES + 2 * h;

  const unsigned lds0 = (unsigned)(uintptr_t)&bs[0][0];
  const unsigned lds1 = (unsigned)(uintptr_t)&bs[1][0];

  // prologue: blocks 0 and 1 in flight
  async_copy_bblk(lds0, B + 0 * HID, lane);
  async_copy_bblk(lds1, B + KB * HID, lane);

  v8f acc = {};
  for (int kb = 0; kb < N_NODES; kb += KB) {
    const int      bufsel = (kb >> 6) & 1;
    const float*   lb     = bufsel ? &bs[1][0] : &bs[0][0];
    const unsigned ldsb   = bufsel ? lds1 : lds0;

    if (kb + 4 * KB < N_NODES)
      __builtin_prefetch(arow + kb + 4 * KB, 0, 1);   // global_prefetch_b8

    // Issue all A fragments for this block (16 x b64) before consuming.
    v2f a[16];
    #pragma unroll
    for (int t = 0; t < 16; ++t)
      a[t] = *(const v2f*)(arow + kb + 4 * t);

    // Oldest async block complete? (8 ops of the next block may remain)
    if (kb + KB < N_NODES)
      asm volatile("s_wait_asynccnt 0x8" ::: "memory");
    else
      asm volatile("s_wait_asynccnt 0x0" ::: "memory");

    #pragma unroll
    for (int t = 0; t < 16; ++t) {
      // B frag (4x16): VGPR0 = rows {4t, 4t+2}, VGPR1 = rows {4t+1, 4t+3};
      // bank-conflict-free: lanes hit banks {m,16+m} and {32+m,48+m}.
      v2f b;
      b.x = lb[(4 * t + 2 * h) * HID + m];
      b.y = lb[(4 * t + 2 * h + 1) * HID + m];
      acc = __builtin_amdgcn_wmma_f32_16x16x4_f32(
          false, a[t], false, b, (short)0, acc, false, false);
    }

    // Refill this buffer with block kb+2 after our LDS reads retire.
    if (kb + 2 * KB < N_NODES) {
      asm volatile("s_wait_dscnt 0x0" ::: "memory");
      async_copy_bblk(ldsb, B + (kb + 2 * KB) * HID, lane);
    }
  }

  // D layout: VGPR v, lane -> M = v + 8*h, N = m (coalesced 64B row stores)
  #pragma unroll
  for (int v = 0; v < 8; ++v)
    C[(long)(rb + v + 8 * h) * HID + m] = acc[v];
}

// ---------------------------------------------------------------------------
// Small WMMA GEMM (layer 2): C[N x 16] = A[N x 16] @ B[16 x 16].
// ---------------------------------------------------------------------------
__global__ __launch_bounds__(32) void k_gemm16(const float* __restrict__ A,
                                               const float* __restrict__ B,
                                               float* __restrict__ C) {
  const int rb   = blockIdx.x * 16;
  const int lane = threadIdx.x;
  const int m    = lane & 15;
  const int h    = lane >> 4;
  const float* __restrict__ arow = A + (long)(rb + m) * HID + 2 * h;
  const float* __restrict__ bcol = B + 2 * h * HID + m;

  v8f acc = {};
  #pragma unroll
  for (int k = 0; k < HID; k += 4) {
    v2f a = *(const v2f*)(arow + k);
    v2f b;
    b.x = bcol[k * HID];
    b.y = bcol[k * HID + HID];
    acc = __builtin_amdgcn_wmma_f32_16x16x4_f32(
        false, a, false, b, (short)0, acc, false, false);
  }
  #pragma unroll
  for (int v = 0; v < 8; ++v)
    C[(long)(rb + v + 8 * h) * HID + m] = acc[v];
}

// ---------------------------------------------------------------------------
// Graph / degree kernels
// ---------------------------------------------------------------------------
__global__ void k_deg_init(float* deg) {            // self-loop contributes 1
  int i = blockIdx.x * blockDim.x + threadIdx.x;
  if (i < N_NODES) deg[i] = 1.0f;
}

__global__ void k_pool_init(float* pooled) {
  int t = threadIdx.x;
  if (t < HID) pooled[t] = 0.0f;
}

__global__ void k_deg_count(const long long* __restrict__ dst, float* deg) {
  int e = blockIdx.x * blockDim.x + threadIdx.x;
  if (e < N_EDGES) atomicAdd(&deg[dst[e]], 1.0f);
}

__global__ void k_deg_rsqrt(float* deg) {
  int i = blockIdx.x * blockDim.x + threadIdx.x;
  if (i < N_NODES) deg[i] = rsqrtf(fmaxf(deg[i], 1e-12f));
}

// out[i] = tmp[i] * dinv[i]^2  (self-loop term; also initializes `out`)
__global__ void k_selfnorm(const float* __restrict__ tmp,
                           const float* __restrict__ dinv,
                           float* __restrict__ out) {
  long g = (long)blockIdx.x * blockDim.x + threadIdx.x;  // N*HID threads
  long i = g >> 4;
  float di = dinv[i];
  out[g] = tmp[g] * di * di;
}

// scatter-add of normalized messages over edges
__global__ void k_edges(const float* __restrict__ tmp,
                        const long long* __restrict__ src,
                        const long long* __restrict__ dst,
                        const float* __restrict__ dinv,
                        float* __restrict__ out) {
  long g = (long)blockIdx.x * blockDim.x + threadIdx.x;  // E*HID threads
  long e = g >> 4;
  int  j = (int)(g & 15);
  long s = (long)src[e];
  long d = (long)dst[e];
  float w = dinv[s] * dinv[d];
  atomicAdd(&out[d * HID + j], tmp[s * HID + j] * w);
}

__global__ void k_bias_relu(float* __restrict__ out, const float* __restrict__ b) {
  long g = (long)blockIdx.x * blockDim.x + threadIdx.x;  // N*HID threads
  out[g] = fmaxf(out[g] + b[g & 15], 0.0f);
}

// ---------------------------------------------------------------------------
// Heads
// ---------------------------------------------------------------------------
__global__ void k_critic(const float* __restrict__ h,
                         const float* __restrict__ cW,
                         const float* __restrict__ cb,
                         float* __restrict__ out) {
  int i = blockIdx.x * blockDim.x + threadIdx.x;
  if (i >= N_NODES) return;
  float s = cb[0];
  #pragma unroll
  for (int j = 0; j < HID; ++j) s += h[(long)i * HID + j] * cW[j];
  out[i] = s;
}

// 64 blocks x 256 threads; block covers 256 rows; 16 atomics per block
__global__ void k_pool_reduce(const float* __restrict__ h, float* pooled) {
  __shared__ float red[256];
  int t = threadIdx.x;
  int r = t >> 4, j = t & 15;
  long base = (long)blockIdx.x * 256;
  float s = 0.0f;
  #pragma unroll
  for (int it = 0; it < 16; ++it)
    s += h[(base + r + it * 16) * HID + j];
  red[t] = s;
  __syncthreads();
  for (int stride = 128; stride >= 16; stride >>= 1) {
    if (t < stride) red[t] += red[t + stride];
    __syncthreads();
  }
  if (t < 16) atomicAdd(&pooled[t], red[t]);
}

// 1 block, 128 threads: pooled/N @ fc_W + fc_b, pairwise softmax*(N-1), round
__global__ void k_actor_head(const float* __restrict__ pooled,
                             const float* __restrict__ fc_W,
                             const float* __restrict__ fc_b,
                             float* __restrict__ out) {
  __shared__ float p[HID];
  __shared__ float o[128];
  int t = threadIdx.x;
  if (t < HID) p[t] = pooled[t] * (1.0f / (float)N_NODES);
  __syncthreads();
  float v = fc_b[t];
  #pragma unroll
  for (int k = 0; k < HID; ++k) v += p[k] * fc_W[k * 128 + t];
  o[t] = v;
  __syncthreads();
  if (t < 64) {
    float a = o[2 * t], b = o[2 * t + 1];
    float mx = fmaxf(a, b);
    float ea = expf(a - mx), eb = expf(b - mx);
    float inv = (float)(N_NODES - 1) / (ea + eb);
    out[2 * t]     = rintf(ea * inv);   // round-half-even == jnp.round
    out[2 * t + 1] = rintf(eb * inv);
  }
}

// ---------------------------------------------------------------------------
extern "C" void kernel_launch(void* const* d_in, const int* in_sizes, int n_in,
                              void* d_out, int out_size, void* d_ws, size_t ws_size,
                              hipStream_t stream) {
  const float*     x    = (const float*)d_in[0];
  const long long* eidx = (const long long*)d_in[1];   // int64 [2, E]
  const float*     W1   = (const float*)d_in[2];
  const float*     b1   = (const float*)d_in[3];
  const float*     W2   = (const float*)d_in[4];
  const float*     b2   = (const float*)d_in[5];
  const float*     fcW  = (const float*)d_in[6];
  const float*     fcb  = (const float*)d_in[7];
  const float*     cW   = (const float*)d_in[8];
  const float*     cb   = (const float*)d_in[9];
  const long long* src  = eidx;
  const long long* dst  = eidx + N_EDGES;

  float* out = (float*)d_out;      // [0..127]=nodes_chosen, [128..]=critic

  // workspace layout
  float* bufA   = (float*)d_ws;                 // N*HID
  float* bufB   = bufA + (long)N_NODES * HID;   // N*HID
  float* dinv   = bufB + (long)N_NODES * HID;   // N
  float* pooled = dinv + N_NODES;               // HID

  const int T = 256;
  dim3 blkT(T);
  dim3 gN((N_NODES + T - 1) / T);
  dim3 gNH((N_NODES * HID) / T);
  dim3 gE((N_EDGES + T - 1) / T);
  dim3 gEH(((long)N_EDGES * HID) / T);
  dim3 gTile(N_NODES / 16);

  // degree -> dinv (recomputed every call; no cross-call state)
  k_deg_init<<<gN, blkT, 0, stream>>>(dinv);
  k_pool_init<<<1, 32, 0, stream>>>(pooled);
  k_deg_count<<<gE, blkT, 0, stream>>>(dst, dinv);
  k_deg_rsqrt<<<gN, blkT, 0, stream>>>(dinv);

  // layer 1: tmp = x @ W1 ; aggregate ; +b1 ; relu
  k_gemm_x<<<gTile, 32, 0, stream>>>(x, W1, bufA);
  k_selfnorm<<<gNH, blkT, 0, stream>>>(bufA, dinv, bufB);
  k_edges<<<gEH, blkT, 0, stream>>>(bufA, src, dst, dinv, bufB);
  k_bias_relu<<<gNH, blkT, 0, stream>>>(bufB, b1);

  // layer 2: tmp = h1 @ W2 ; aggregate ; +b2 ; relu
  k_gemm16<<<gTile, 32, 0, stream>>>(bufB, W2, bufA);
  k_selfnorm<<<gNH, blkT, 0, stream>>>(bufA, dinv, bufB);
  k_edges<<<gEH, blkT, 0, stream>>>(bufA, src, dst, dinv, bufB);
  k_bias_relu<<<gNH, blkT, 0, stream>>>(bufB, b2);

  // heads
  k_critic<<<gN, blkT, 0, stream>>>(bufB, cW, cb, out + 128);
  k_pool_reduce<<<64, blkT, 0, stream>>>(bufB, pooled);
  k_actor_head<<<1, 128, 0, stream>>>(pooled, fcW, fcb, out);
}